// CLAHEEffect_41549513621550
// MI455X (gfx1250) — compile-verified
//
#include <hip/hip_runtime.h>

// ---------------- problem constants ----------------
#define H       4096
#define W       4096
#define PLANE   (H*W)              // 16777216
#define TDIM    512                // tile edge (h//TILE_H)
#define NTILES  64                 // 8x8 tiles
#define NB      256                // histogram bins
#define TPIX    (TDIM*TDIM)        // 262144 pixels per tile
#define SUBS    32                 // blocks per tile for streaming passes
#define PXB     (TPIX/SUBS)        // 8192 pixels per block
#define CLIPLIM 4.0f
#define CONTRAST 1.15f
#define BLEND   0.65f
#define FIXSCALE 16777216.0f       // 2^24 fixed point for deterministic mean

typedef float f4  __attribute__((ext_vector_type(4)));
typedef float v2f __attribute__((ext_vector_type(2)));
typedef float v8f __attribute__((ext_vector_type(8)));

// ---------------- small per-tile state (device globals: no ws-size risk) ----
__device__ unsigned int       g_hist[NTILES*NB];
__device__ unsigned int       g_minb[NTILES];
__device__ unsigned int       g_maxb[NTILES];
__device__ float              g_lut [NTILES*NB];
__device__ unsigned long long g_eqsum[NTILES];

// f32 WMMA: D(16x16) = A(16x4) * B(4x16) + C   (exact for our integer-count sums)
__device__ __forceinline__ v8f wmma_f32(v2f a, v2f b, v8f c) {
  return __builtin_amdgcn_wmma_f32_16x16x4_f32(
      /*neg_a=*/false, a, /*neg_b=*/false, b,
      /*c_mod=*/(short)0, c, /*reuse_a=*/false, /*reuse_b=*/false);
}

// ---------------- kernel 0: init per-tile state ----------------
__global__ void __launch_bounds__(256) k_init() {
  int i = blockIdx.x * 256 + threadIdx.x;
  if (i < NTILES*NB) g_hist[i] = 0u;
  if (i < NTILES) {
    g_minb[i]  = 0x7F800000u;   // +inf bits (positive floats: uint order == float order)
    g_maxb[i]  = 0u;
    g_eqsum[i] = 0ull;
  }
}

// ---------------- kernel 1: per-tile histogram + min/max (RT loads: prime L2)
__global__ void __launch_bounds__(256) k_stats(const float* __restrict__ img) {
  const int tile = blockIdx.x >> 5;
  const int sub  = blockIdx.x & 31;
  const int tid  = threadIdx.x;
  __shared__ unsigned int s_hist[NB];
  __shared__ unsigned int s_min, s_max;
  s_hist[tid] = 0u;
  if (tid == 0) { s_min = 0x7F800000u; s_max = 0u; }
  __syncthreads();

  const int ti = tile >> 3, tj = tile & 7;
  const int tilebase = (ti*TDIM)*W + tj*TDIM;
  float vmin = __uint_as_float(0x7F800000u), vmax = 0.0f;

  for (int k = 0; k < 8; ++k) {
    const int p = sub*PXB + k*1024 + tid*4;
    const int off = tilebase + (p >> 9)*W + (p & 511);
    if (k < 7) { // prefetch next chunk of all 3 planes -> global_prefetch_b8
      const int pn = p + 1024;
      const int offn = tilebase + (pn >> 9)*W + (pn & 511);
      __builtin_prefetch(img + offn);
      __builtin_prefetch(img + PLANE + offn);
      __builtin_prefetch(img + 2*PLANE + offn);
    }
    f4 R = *(const f4*)(img + off);
    f4 G = *(const f4*)(img + PLANE + off);
    f4 B = *(const f4*)(img + 2*PLANE + off);
    #pragma unroll
    for (int q = 0; q < 4; ++q) {
      float y = 0.299f*R[q] + 0.587f*G[q] + 0.114f*B[q];
      int bin = (int)(y * 256.0f);
      bin = bin < 0 ? 0 : (bin > 255 ? 255 : bin);
      atomicAdd(&s_hist[bin], 1u);
      vmin = fminf(vmin, y);
      vmax = fmaxf(vmax, y);
    }
  }
  atomicMin(&s_min, __float_as_uint(vmin));
  atomicMax(&s_max, __float_as_uint(vmax));
  __syncthreads();
  unsigned int hv = s_hist[tid];
  if (hv) atomicAdd(&g_hist[tile*NB + tid], hv);
  if (tid == 0) {
    atomicMin(&g_minb[tile], s_min);
    atomicMax(&g_maxb[tile], s_max);
  }
}

// ---------------- kernel 2: clip + redistribute + cumsum (WMMA) -> LUT ------
// One wave32 per tile. 256 bins as 16x16 matrix M[i][j] = h[16i+j].
//   RowPrefix = M x U            (U upper-tri ones, K=16 split into 4 WMMA k-chunks)
//   CDF       = RowPrefix + Lstrict x (t * 1^T)   (t = row sums, 4 more k-chunks)
// Layouts per CDNA5 ISA (32-bit A 16x4: lanes0-15 K0/K1, lanes16-31 K2/K3;
// B/C/D rows striped across lanes within a VGPR).
__global__ void __launch_bounds__(32) k_lut() {
  const int tile = blockIdx.x;
  const int l    = threadIdx.x;
  const bool hi  = (l >= 16);
  const int  m   = l & 15;
  __shared__ float sh[NB];
  __shared__ float st[16];

  float h[8];
  float pe = 0.0f;
  #pragma unroll
  for (int k = 0; k < 8; ++k) {
    h[k] = (float)g_hist[tile*NB + l*8 + k];
    pe  += fmaxf(h[k] - CLIPLIM, 0.0f);
  }
  #pragma unroll
  for (int s = 16; s >= 1; s >>= 1) pe += __shfl_xor(pe, s, 32);   // wave32 reduce
  const float add = pe * (1.0f/256.0f);
  #pragma unroll
  for (int k = 0; k < 8; ++k) sh[l*8 + k] = fminf(h[k], CLIPLIM) + add;
  __syncthreads();

  v8f acc = {0.f,0.f,0.f,0.f,0.f,0.f,0.f,0.f};
  #pragma unroll
  for (int kc = 0; kc < 4; ++kc) {            // RowPrefix accumulation
    v2f a, b;
    const int k0 = 4*kc + (hi ? 2 : 0);
    a.x = sh[16*m + k0];
    a.y = sh[16*m + k0 + 1];
    b.x = ((4*kc + (hi ? 1 : 0)) <= m) ? 1.0f : 0.0f;   // U[k][j] = (k<=j)
    b.y = ((4*kc + (hi ? 3 : 2)) <= m) ? 1.0f : 0.0f;
    acc = wmma_f32(a, b, acc);
  }
  // extract row totals t[i] = RowPrefix[i][15]
  if (l == 15) {
    #pragma unroll
    for (int v = 0; v < 8; ++v) st[v] = acc[v];
  }
  if (l == 31) {
    #pragma unroll
    for (int v = 0; v < 8; ++v) st[8 + v] = acc[v];
  }
  __syncthreads();
  #pragma unroll
  for (int kc = 0; kc < 4; ++kc) {            // + Lstrict x (t broadcast)
    v2f a, b;
    a.x = ((4*kc + (hi ? 2 : 0)) < m) ? 1.0f : 0.0f;    // Lstrict[i][k] = (k<i)
    a.y = ((4*kc + (hi ? 3 : 1)) < m) ? 1.0f : 0.0f;
    b.x = st[4*kc + (hi ? 1 : 0)];                      // row-constant t
    b.y = st[4*kc + (hi ? 3 : 2)];
    acc = wmma_f32(a, b, acc);
  }
  const float last  = __shfl(acc[7], 31, 32);           // cdf[255]
  const float scale = (last > 0.0f) ? (1.0f/last) : 1.0f;
  #pragma unroll
  for (int v = 0; v < 8; ++v) {
    const int bin = 16*(v + (hi ? 8 : 0)) + m;
    g_lut[tile*NB + bin] = acc[v] * scale;
  }
}

// ---------------- kernel 3: per-tile mean(eq) (deterministic u64 fixed-point)
__global__ void __launch_bounds__(256) k_eqsum(const float* __restrict__ img) {
  const int tile = blockIdx.x >> 5;
  const int sub  = blockIdx.x & 31;
  const int tid  = threadIdx.x;
  __shared__ float s_lut[NB];
  __shared__ unsigned long long s_sum;
  s_lut[tid] = g_lut[tile*NB + tid];
  if (tid == 0) s_sum = 0ull;
  const float tmin = __uint_as_float(g_minb[tile]);
  const float tmax = __uint_as_float(g_maxb[tile]);
  __syncthreads();
  if (!(tmax > tmin)) return;                 // invalid tile: mean unused
  const float inv = 1.0f / (tmax - tmin);

  const int ti = tile >> 3, tj = tile & 7;
  const int tilebase = (ti*TDIM)*W + tj*TDIM;
  unsigned int accu = 0u;                     // <= 32 * 2^24 < 2^32
  for (int k = 0; k < 8; ++k) {
    const int p = sub*PXB + k*1024 + tid*4;
    const int off = tilebase + (p >> 9)*W + (p & 511);
    f4 R = *(const f4*)(img + off);           // expect L2 hits (primed by pass 1)
    f4 G = *(const f4*)(img + PLANE + off);
    f4 B = *(const f4*)(img + 2*PLANE + off);
    #pragma unroll
    for (int q = 0; q < 4; ++q) {
      float y = 0.299f*R[q] + 0.587f*G[q] + 0.114f*B[q];
      float norm = (y - tmin) * inv;
      int idx = (int)(norm * 255.0f);
      idx = idx < 0 ? 0 : (idx > 255 ? 255 : idx);
      accu += (unsigned int)(s_lut[idx] * FIXSCALE + 0.5f);
    }
  }
  atomicAdd(&s_sum, (unsigned long long)accu);
  __syncthreads();
  if (tid == 0) atomicAdd(&g_eqsum[tile], s_sum);
}

// ---------------- kernel 4: apply (NT loads for last read, NT stores out) ---
__global__ void __launch_bounds__(256) k_apply(const float* __restrict__ img,
                                               float* __restrict__ out) {
  const int tile = blockIdx.x >> 5;
  const int sub  = blockIdx.x & 31;
  const int tid  = threadIdx.x;
  __shared__ float s_lut[NB];
  s_lut[tid] = g_lut[tile*NB + tid];
  __syncthreads();
  const float tmin  = __uint_as_float(g_minb[tile]);
  const float tmax  = __uint_as_float(g_maxb[tile]);
  const bool  valid = (tmax > tmin);
  const float inv   = valid ? 1.0f / (tmax - tmin) : 0.0f;
  const float mean  = (float)g_eqsum[tile] * (1.0f / (FIXSCALE * (float)TPIX));

  const int ti = tile >> 3, tj = tile & 7;
  const int tilebase = (ti*TDIM)*W + tj*TDIM;
  for (int k = 0; k < 8; ++k) {
    const int p = sub*PXB + k*1024 + tid*4;
    const int off = tilebase + (p >> 9)*W + (p & 511);
    f4 R = __builtin_nontemporal_load((const f4*)(img + off));          // last use
    f4 G = __builtin_nontemporal_load((const f4*)(img + PLANE + off));
    f4 B = __builtin_nontemporal_load((const f4*)(img + 2*PLANE + off));
    f4 oR, oG, oB;
    #pragma unroll
    for (int q = 0; q < 4; ++q) {
      float y = 0.299f*R[q] + 0.587f*G[q] + 0.114f*B[q];
      float ynew = y;
      if (valid) {
        float norm = (y - tmin) * inv;
        int idx = (int)(norm * 255.0f);
        idx = idx < 0 ? 0 : (idx > 255 ? 255 : idx);
        float eq = s_lut[idx];
        eq = (eq - mean) * CONTRAST + mean;
        eq = fminf(fmaxf(eq, 0.0f), 1.0f);
        ynew = y + BLEND * (eq - y);
      }
      const float ratio = (y > 0.01f) ? (ynew / y) : 1.0f;
      oR[q] = fminf(fmaxf(R[q] * ratio, 0.0f), 1.0f);
      oG[q] = fminf(fmaxf(G[q] * ratio, 0.0f), 1.0f);
      oB[q] = fminf(fmaxf(B[q] * ratio, 0.0f), 1.0f);
    }
    __builtin_nontemporal_store(oR, (f4*)(out + off));                  // stream out
    __builtin_nontemporal_store(oG, (f4*)(out + PLANE + off));
    __builtin_nontemporal_store(oB, (f4*)(out + 2*PLANE + off));
  }
}

// ---------------- host entry ----------------
extern "C" void kernel_launch(void* const* d_in, const int* in_sizes, int n_in,
                              void* d_out, int out_size, void* d_ws, size_t ws_size,
                              hipStream_t stream) {
  (void)in_sizes; (void)n_in; (void)out_size; (void)d_ws; (void)ws_size;
  const float* img = (const float*)d_in[0];
  float* out = (float*)d_out;

  k_init  <<<(NTILES*NB + 255)/256, 256, 0, stream>>>();
  k_stats <<<NTILES*SUBS, 256, 0, stream>>>(img);
  k_lut   <<<NTILES, 32, 0, stream>>>();
  k_eqsum <<<NTILES*SUBS, 256, 0, stream>>>(img);
  k_apply <<<NTILES*SUBS, 256, 0, stream>>>(img, out);
}